// SO3onB3_63866163692180
// MI455X (gfx1250) — compile-verified
//
#include <hip/hip_runtime.h>
#include <math.h>

#define LMAXX 5
#define NLAT 12
#define NLON 24
#define NR   32
#define NB   128
#define LL   64
#define NPTS (NR*NLAT*NLON)   /* 9216 */
#define KDIM NPTS
#define NDIM 80               /* 72 valid complex-coeff columns, padded to 80 */
#define NOUT 69

typedef __attribute__((ext_vector_type(2))) float v2f;
typedef __attribute__((ext_vector_type(8))) float v8f;

// ---- workspace layout (units: floats) -------------------------------------
constexpr size_t OFF_RADII  = 0;        //   32
constexpr size_t OFF_RW     = 64;       //  192  rw[l*32+r] = qw*r^(l+2)
constexpr size_t OFF_WP     = 256;      //  432  wp[(l*6+m)*12+j] = Pbar*ccw
constexpr size_t OFF_TRIGC  = 704;      //  144  cos(m*phi)*2pi/24
constexpr size_t OFF_TRIGS  = 848;      //  144  -sin(m*phi)*2pi/24
constexpr size_t OFF_IDX3   = 1024;     //  207 ints (l1,l2,l) per output
constexpr size_t OFF_CG     = 1280;     //  69*1331 = 91839
constexpr size_t OFF_W      = 93184;    //  9216*80 = 737280
constexpr size_t OFF_METAI  = 830464;   //  int4  x 9216
constexpr size_t OFF_METAF  = 867328;   //  float4 x 9216
constexpr size_t OFF_SHELLS = 904192;   //  128*9216 = 1179648
constexpr size_t OFF_AOUT   = 2083840;  //  128*80  -> total ~8.4 MB

__device__ double dfact(int n){ double r = 1.0; for(int i = 2; i <= n; ++i) r *= (double)i; return r; }

__device__ double cg_coef(int j1,int m1,int j2,int m2,int j,int m){
  if (m1+m2 != m || j < abs(j1-j2) || j > j1+j2) return 0.0;
  double pre = sqrt((2.0*j+1.0)*dfact(j+j1-j2)*dfact(j-j1+j2)*dfact(j1+j2-j)/dfact(j1+j2+j+1));
  pre *= sqrt(dfact(j+m)*dfact(j-m)*dfact(j1-m1)*dfact(j1+m1)*dfact(j2-m2)*dfact(j2+m2));
  double s = 0.0;
  for(int k = 0; k <= j1+j2-j; ++k){
    int d0=k, d1=j1+j2-j-k, d2=j1-m1-k, d3=j2+m2-k, d4=j-j2+m1+k, d5=j-j1-m2+k;
    if(d0<0||d1<0||d2<0||d3<0||d4<0||d5<0) continue;
    double t = 1.0/(dfact(d0)*dfact(d1)*dfact(d2)*dfact(d3)*dfact(d4)*dfact(d5));
    s += (k & 1) ? -t : t;
  }
  return pre*s;
}

// ---- kernel 1: all small constant tables (every job independent, no barriers)
__global__ void setup_tables(float* __restrict__ ws){
  const int tid = threadIdx.x;
  float* radii = ws + OFF_RADII;
  float* rw    = ws + OFF_RW;
  float* wp    = ws + OFF_WP;
  float* trigc = ws + OFF_TRIGC;
  float* trigs = ws + OFF_TRIGS;
  int*   idx3  = (int*)(ws + OFF_IDX3);
  float* cg    = ws + OFF_CG;
  const double PI = 3.14159265358979323846;

  // Gauss-Legendre nodes/weights on [0,1] via Newton (one node per thread)
  if (tid < NR){
    double x = cos(PI*(tid + 0.75)/(NR + 0.5));
    double dp = 1.0;
    for(int it = 0; it < 60; ++it){
      double p0 = 1.0, p1 = x;
      for(int n = 2; n <= NR; ++n){ double pn = ((2.0*n-1.0)*x*p1-(n-1.0)*p0)/n; p0 = p1; p1 = pn; }
      dp = NR*(x*p1 - p0)/(x*x - 1.0);
      x -= p1/dp;
    }
    double w  = 2.0/((1.0 - x*x)*dp*dp);
    double r  = (x + 1.0)*0.5;
    double wr = w*0.5;
    radii[tid] = (float)r;
    double pw = r*r;                               // r^(l+2), l=0
    for(int l = 0; l <= LMAXX; ++l){ rw[l*NR+tid] = (float)(wr*pw); pw *= r; }
  }
  // Clenshaw-Curtis weights + normalized associated Legendre (one latitude per thread)
  if (tid >= 32 && tid < 32 + NLAT){
    const int j = tid - 32;
    double th = (double)j*PI/(NLAT-1);
    double w = 1.0; const int Kh = (NLAT-1)/2;
    for(int k = 1; k <= Kh; ++k){
      double b = (2*k == NLAT-1) ? 1.0 : 2.0;
      w -= b*cos(2.0*k*th)/(4.0*k*k - 1.0);
    }
    w *= 2.0/(NLAT-1);
    if (j == 0 || j == NLAT-1) w *= 0.5;
    double x = cos(th), s = sqrt(fmax(0.0, 1.0 - x*x));
    double P[LMAXX+1][LMAXX+1];
    for(int a = 0; a <= LMAXX; ++a) for(int b = 0; b <= LMAXX; ++b) P[a][b] = 0.0;
    P[0][0] = sqrt(1.0/(4.0*PI));
    for(int m = 1; m <= LMAXX; ++m) P[m][m] = -sqrt((2.0*m+1.0)/(2.0*m))*s*P[m-1][m-1];
    for(int m = 0; m <  LMAXX; ++m) P[m+1][m] = sqrt(2.0*m+3.0)*x*P[m][m];
    for(int m = 0; m <= LMAXX; ++m)
      for(int l = m+2; l <= LMAXX; ++l){
        double a = sqrt((4.0*l*l-1.0)/(double)(l*l - m*m));
        double b = sqrt(((l-1.0)*(l-1.0)-m*m)/(4.0*(l-1.0)*(l-1.0)-1.0));
        P[l][m] = a*(x*P[l-1][m] - b*P[l-2][m]);
      }
    for(int l = 0; l <= LMAXX; ++l)
      for(int m = 0; m <= LMAXX; ++m)
        wp[(l*6+m)*NLAT + j] = (float)(P[l][m]*w);   // m>l stays 0
  }
  // DFT twiddles (fft sign convention: e^{-i m phi}), scaled by 2pi/NLON
  if (tid >= 64 && tid < 64 + 6*NLON){
    int q = tid - 64, m = q/NLON, i = q % NLON;
    double ph = 2.0*PI*(double)i/NLON, sc = 2.0*PI/NLON;
    trigc[q] = (float)( cos(m*ph)*sc);
    trigs[q] = (float)(-sin(m*ph)*sc);
  }
  // Clebsch-Gordan tensors + (l1,l2,l) index map (one output channel per thread)
  if (tid < NOUT){
    int cnt = 0, L1 = 0, L2 = 0, LO = 0;
    for(int l1 = 0; l1 <= LMAXX; ++l1)
      for(int l2 = l1; l2 <= LMAXX; ++l2){
        int lhi = (l1+l2 < LMAXX) ? (l1+l2) : LMAXX;
        for(int l = l2-l1; l <= lhi; ++l){ if (cnt == tid){ L1=l1; L2=l2; LO=l; } ++cnt; }
      }
    idx3[tid*3+0] = L1; idx3[tid*3+1] = L2; idx3[tid*3+2] = LO;
    float* T = cg + (size_t)tid*1331;
    for(int e = 0; e < 1331; ++e) T[e] = 0.f;
    for(int m1 = -L1; m1 <= L1; ++m1)
      for(int m2 = -L2; m2 <= L2; ++m2){
        int m = m1 + m2;
        if (abs(m) <= LO)
          T[(m1+L1)*121 + (m2+L2)*11 + (m+LO)] = (float)cg_coef(L1,m1,L2,m2,LO,m);
      }
  }
}

// ---- kernel 2: fused spectral matrix W[9216][80] + trilinear sample metadata
__global__ void setup_grid(float* __restrict__ ws){
  const int k = blockIdx.x*blockDim.x + threadIdx.x;
  if (k >= KDIM) return;
  const float* radii = ws + OFF_RADII;
  const float* rw    = ws + OFF_RW;
  const float* wp    = ws + OFF_WP;
  const float* trigc = ws + OFF_TRIGC;
  const float* trigs = ws + OFF_TRIGS;
  const double PI = 3.14159265358979323846;

  const int i = k % NLON, j = (k/NLON) % NLAT, r = k/(NLON*NLAT);
  float* Wrow = ws + OFF_W + (size_t)k*NDIM;
  for(int n = 0; n < NDIM; ++n){
    float v = 0.f;
    if (n < 72){
      int c = n & 1, lm = n >> 1, l = lm/6, m = lm % 6;
      if (m <= l){
        float t = c ? trigs[m*NLON+i] : trigc[m*NLON+i];
        v = rw[l*NR+r]*wp[(l*6+m)*NLAT+j]*t;
      }
    }
    Wrow[n] = v;
  }
  // trilinear metadata for this grid point
  double th = (double)j*PI/(NLAT-1);
  double ph = 2.0*PI*(double)i/NLON;
  double rr = (double)radii[r];
  float cx = (float)(rr*sin(th)*cos(ph));
  float cy = (float)(rr*sin(th)*sin(ph));
  float cz = (float)(rr*cos(th));
  const float sc = (LL-1)*0.5f;
  float ix = (cx+1.f)*sc, iy = (cy+1.f)*sc, iz = (cz+1.f)*sc;
  int x0 = (int)floorf(ix), y0 = (int)floorf(iy), z0 = (int)floorf(iz);
  x0 = min(max(x0,0), LL-2); y0 = min(max(y0,0), LL-2); z0 = min(max(z0,0), LL-2);
  int4*   mi = (int4*)  (ws + OFF_METAI);
  float4* mf = (float4*)(ws + OFF_METAF);
  mi[k] = make_int4(x0, y0, z0, 0);
  mf[k] = make_float4(ix - (float)x0, iy - (float)y0, iz - (float)z0, 0.f);
}

// ---- kernel 3: the only HBM-heavy step: gather + trilinear blend ----------
__global__ void __launch_bounds__(256) sample_kernel(const float* __restrict__ f,
                                                     float* __restrict__ ws){
  const int gid = blockIdx.x*blockDim.x + threadIdx.x;
  if (gid >= NB*NPTS) return;
  const int p = gid % NPTS, b = gid / NPTS;
  const int4   c = ((const int4*)  (ws + OFF_METAI))[p];
  const float4 t = ((const float4*)(ws + OFF_METAF))[p];
  const float* fb = f + ((size_t)b << 18);            // 64^3 per batch
  const size_t base = ((size_t)c.z << 12) + ((size_t)c.y << 6) + (size_t)c.x;
  float v000 = fb[base],        v001 = fb[base+1];
  float v010 = fb[base+64],     v011 = fb[base+65];
  float v100 = fb[base+4096],   v101 = fb[base+4097];
  float v110 = fb[base+4160],   v111 = fb[base+4161];
  float wx1 = t.x, wx0 = 1.f-t.x, wy1 = t.y, wy0 = 1.f-t.y, wz1 = t.z, wz0 = 1.f-t.z;
  float v = wz0*(wy0*(wx0*v000 + wx1*v001) + wy1*(wx0*v010 + wx1*v011))
          + wz1*(wy0*(wx0*v100 + wx1*v101) + wy1*(wx0*v110 + wx1*v111));
  ws[OFF_SHELLS + (size_t)b*KDIM + p] = v;
}

// ---- kernel 4: A = shells(128x9216) * W(9216x80) via V_WMMA_F32_16X16X4_F32
// 8 M-tiles x 5 N-tiles; 4 waves/block split K, deterministic LDS reduction.
__global__ void __launch_bounds__(128) gemm_wmma(float* __restrict__ ws){
  const float* S = ws + OFF_SHELLS;   // [128][9216] row-major
  const float* W = ws + OFF_W;        // [9216][80]  row-major
  float* A       = ws + OFF_AOUT;     // [128][80]
  __shared__ float red[4][16][16];

  const int tid  = threadIdx.x;
  const int wave = tid >> 5, lane = tid & 31;
  const int row  = lane & 15, hi = lane >> 4;
  const int mt   = blockIdx.x / 5, nt = blockIdx.x % 5;
  const int m    = mt*16 + row;       // A-matrix row for this lane
  const int n    = nt*16 + row;       // B-matrix column for this lane

  const float* srow = S + (size_t)m*KDIM;
  v8f acc = {};
  const int kbeg = wave*(KDIM/4), kend = kbeg + (KDIM/4);
  for(int k = kbeg; k < kend; k += 4){
    // A 16x4 f32 layout: VGPR0/1 hold K={0,1} (lanes 0-15) / K={2,3} (lanes 16-31)
    v2f a = *(const v2f*)(srow + k + 2*hi);
    // B 4x16 f32 layout: VGPR0 = rows K=0/K=2 (lane halves), VGPR1 = rows K=1/K=3
    v2f b;
    b[0] = W[(size_t)(k + 2*hi    )*NDIM + n];
    b[1] = W[(size_t)(k + 2*hi + 1)*NDIM + n];
    acc = __builtin_amdgcn_wmma_f32_16x16x4_f32(false, a, false, b,
                                                (short)0, acc, false, false);
  }
  // C/D layout: VGPR v -> row v (lanes 0-15) / row v+8 (lanes 16-31), col = lane&15
  #pragma unroll
  for(int v = 0; v < 8; ++v) red[wave][v + 8*hi][row] = acc[v];
  __syncthreads();
  for(int e = tid; e < 256; e += 128){
    int rr = e >> 4, cc = e & 15;
    float s = red[0][rr][cc] + red[1][rr][cc] + red[2][rr][cc] + red[3][rr][cc];
    A[(size_t)(mt*16 + rr)*NDIM + nt*16 + cc] = s;
  }
}

// ---- kernel 5: conj-symmetric expansion + CG triple contraction -----------
__global__ void __launch_bounds__(128) bispec(const float* __restrict__ ws,
                                              float* __restrict__ out){
  const int b = blockIdx.x, tid = threadIdx.x;
  __shared__ float2 full[6][11];
  const float* A = ws + OFF_AOUT + (size_t)b*NDIM;

  if (tid < 36){
    int l = 0; while ((l+1)*(l+1) <= tid) ++l;     // l^2 <= tid < (l+1)^2
    int idx = tid - l*l;
    int ms = idx - l, mm = ms < 0 ? -ms : ms;
    float re = A[2*(l*6+mm)], im = A[2*(l*6+mm)+1];
    float2 v;
    if (ms >= 0){ v.x = re; v.y = im; }
    else { float sg = (mm & 1) ? -1.f : 1.f; v.x = sg*re; v.y = -sg*im; }
    full[l][idx] = v;
  }
  __syncthreads();
  if (tid < NOUT){
    const int* idx3 = (const int*)(ws + OFF_IDX3);
    const int l1 = idx3[tid*3], l2 = idx3[tid*3+1], l = idx3[tid*3+2];
    const float* T = ws + OFF_CG + (size_t)tid*1331;
    float ar = 0.f, ai = 0.f;
    for(int i = 0; i < 2*l1+1; ++i){
      float2 u = full[l1][i];
      for(int j = 0; j < 2*l2+1; ++j){
        float2 w = full[l2][j];
        float pr = u.x*w.x - u.y*w.y;
        float pi = u.x*w.y + u.y*w.x;
        const float* Trow = T + (i*121 + j*11);
        for(int k = 0; k < 2*l+1; ++k){
          float g = Trow[k];
          float2 z = full[l][k];                    // conj(z) = (z.x, -z.y)
          ar += g*(pr*z.x + pi*z.y);
          ai += g*(pi*z.x - pr*z.y);
        }
      }
    }
    out[((size_t)b*NOUT + tid)*2    ] = ar;
    out[((size_t)b*NOUT + tid)*2 + 1] = ai;
  }
}

extern "C" void kernel_launch(void* const* d_in, const int* in_sizes, int n_in,
                              void* d_out, int out_size, void* d_ws, size_t ws_size,
                              hipStream_t stream){
  const float* f = (const float*)d_in[0];
  float* ws  = (float*)d_ws;
  float* out = (float*)d_out;
  (void)in_sizes; (void)n_in; (void)out_size; (void)ws_size;

  setup_tables<<<1, 256, 0, stream>>>(ws);
  setup_grid  <<<(NPTS + 255)/256, 256, 0, stream>>>(ws);
  sample_kernel<<<(NB*NPTS + 255)/256, 256, 0, stream>>>(f, ws);
  gemm_wmma   <<<40, 128, 0, stream>>>(ws);
  bispec      <<<NB, 128, 0, stream>>>(ws, out);
}